// Detector_24833500906204
// MI455X (gfx1250) — compile-verified
//
#include <hip/hip_runtime.h>
#include <hip/hip_bf16.h>

// YOLO decode: [B,45,H,W] -> boxes [B*H*W*3, 6] + mask [B*H*W*3]
// Memory-bound (~120 MB moved, ~55 MFLOP -> ~5us at 23.3 TB/s). No matrix op
// exists in this workload, so WMMA is intentionally not used. Optimization is
// pure data movement: compile-time H/W (no integer division), immediate-offset
// coalesced b32 loads, 8-byte NT streaming stores, global_prefetch_b8.

#define NUM_CLASSES 10

// Clang-native vector type: accepted by __builtin_nontemporal_store.
typedef __attribute__((ext_vector_type(2))) float v2f;

template <int H, int W, int STRIDE>
__global__ __launch_bounds__(256) void yolo_decode_kernel(
    const float* __restrict__ out,      // [B, 45, H, W]
    const float* __restrict__ anchors,  // [3, 2]
    const float* __restrict__ thresh_p, // scalar
    float* __restrict__ boxes,          // [Ntot, 6]
    float* __restrict__ mask,           // [Ntot]
    long long slot_base)
{
    constexpr int HW = H * W;

    const int cell = blockIdx.x * blockDim.x + threadIdx.x;
    if (cell >= HW) return;
    const int b = blockIdx.y;

    const int y = cell / W;            // constant divisor -> mulhi+shift
    const int x = cell - y * W;
    const float fx = (float)x;
    const float fy = (float)y;
    const float thresh = *thresh_p;

    // Base of this cell's 45-channel column; channel offsets are immediates.
    const float* __restrict__ base = out + (long long)b * (45 * HW) + cell;

    // gfx1250: global_prefetch_b8 — pull the strided channel column toward cache.
    __builtin_prefetch(base, 0, 1);
    __builtin_prefetch(base + 22 * HW, 0, 1);

    const long long slot0 = slot_base + ((long long)b * HW + cell) * 3;

    #pragma unroll
    for (int a = 0; a < 3; ++a) {
        // All offsets below are compile-time constants -> IOFFSET immediates.
        const float iou = base[(a * 15 + 0) * HW];
        const float dx  = base[(a * 15 + 1) * HW];
        const float dy  = base[(a * 15 + 2) * HW];
        const float dw  = base[(a * 15 + 3) * HW];
        const float dh  = base[(a * 15 + 4) * HW];

        const float pre_x = (fx + dx) * (float)STRIDE;
        const float pre_y = (fy + dy) * (float)STRIDE;

        const float aw = anchors[a * 2 + 0];
        const float ah = anchors[a * 2 + 1];
        const float pw = aw * __expf(dw);   // v_exp_f32 (TRANS, co-executes)
        const float ph = ah * __expf(dh);

        // First-occurrence argmax over 10 class scores (strict > keeps first).
        float best = base[(a * 15 + 5) * HW];
        float kind = 0.0f;
        #pragma unroll
        for (int k = 1; k < NUM_CLASSES; ++k) {
            const float v = base[(a * 15 + 5 + k) * HW];
            if (v > best) { best = v; kind = (float)k; }
        }

        const float hw2 = 0.5f * pw;
        const float hh2 = 0.5f * ph;
        const float x1 = pre_x - hw2;
        const float y1 = pre_y - hh2;
        const float x2 = pre_x + hw2;
        const float y2 = pre_y + hh2;

        const long long s = slot0 + a;
        // Row base is a 24-byte multiple from 256B-aligned d_out -> 8B aligned.
        v2f* __restrict__ bo2 = (v2f*)(boxes + s * 6);

        // Streaming outputs: NT temporal hint (write-once, never re-read).
        __builtin_nontemporal_store((v2f){iou, x1},  bo2 + 0);
        __builtin_nontemporal_store((v2f){y1,  x2},  bo2 + 1);
        __builtin_nontemporal_store((v2f){y2, kind}, bo2 + 2);
        __builtin_nontemporal_store(iou > thresh ? 1.0f : 0.0f, mask + s);
    }
}

extern "C" void kernel_launch(void* const* d_in, const int* in_sizes, int n_in,
                              void* d_out, int out_size, void* d_ws, size_t ws_size,
                              hipStream_t stream) {
    const float* o13 = (const float*)d_in[0];
    const float* o26 = (const float*)d_in[1];
    const float* o52 = (const float*)d_in[2];
    const float* a13 = (const float*)d_in[3];
    const float* a26 = (const float*)d_in[4];
    const float* a52 = (const float*)d_in[5];
    const float* thr = (const float*)d_in[6];

    // Derive batch from input sizes: in_sizes[0] == B*45*13*13.
    const int B = in_sizes[0] / (45 * 13 * 13);

    const long long N13 = (long long)B * 13 * 13 * 3;
    const long long N26 = (long long)B * 26 * 26 * 3;
    const long long N52 = (long long)B * 52 * 52 * 3;
    const long long Ntot = N13 + N26 + N52;

    float* boxes = (float*)d_out;            // [Ntot, 6]
    float* mask  = (float*)d_out + Ntot * 6; // [Ntot]

    const int threads = 256; // 8 wave32s per block

    {
        constexpr int HW = 13 * 13;
        dim3 grid((HW + threads - 1) / threads, B);
        yolo_decode_kernel<13, 13, 32><<<grid, threads, 0, stream>>>(
            o13, a13, thr, boxes, mask, 0LL);
    }
    {
        constexpr int HW = 26 * 26;
        dim3 grid((HW + threads - 1) / threads, B);
        yolo_decode_kernel<26, 26, 16><<<grid, threads, 0, stream>>>(
            o26, a26, thr, boxes, mask, N13);
    }
    {
        constexpr int HW = 52 * 52;
        dim3 grid((HW + threads - 1) / threads, B);
        yolo_decode_kernel<52, 52, 8><<<grid, threads, 0, stream>>>(
            o52, a52, thr, boxes, mask, N13 + N26);
    }
}